// LinEq2v2_42142219108546
// MI455X (gfx1250) — compile-verified
//
#include <hip/hip_runtime.h>

#define NN 2048
#define NO 16      // OUT
#define CP 32      // colsum partitions
#define CR (NN / CP)

typedef __attribute__((ext_vector_type(2))) float v2f;
typedef __attribute__((ext_vector_type(8))) float v8f;

// d_ws float layout:
//  [0,2048)            rowsum
//  [2048,4096)         diag
//  [4096,6144)         colsum
//  [6144..6145]        trace, totsum
//  [8192, 8192+CP*NN)  colsum partials

__global__ void k_row_diag(const float* __restrict__ in,
                           float* __restrict__ rowsum, float* __restrict__ diag) {
  __shared__ float s[256];
  int i = blockIdx.x;
  size_t base = (size_t)i * NN;
  float acc = 0.f;
  for (int k = threadIdx.x; k < NN; k += 256) acc += in[base + k];
  s[threadIdx.x] = acc;
  __syncthreads();
  for (int off = 128; off > 0; off >>= 1) {
    if ((int)threadIdx.x < off) s[threadIdx.x] += s[threadIdx.x + off];
    __syncthreads();
  }
  if (threadIdx.x == 0) { rowsum[i] = s[0]; diag[i] = in[base + i]; }
}

__global__ void k_col_part(const float* __restrict__ in, float* __restrict__ part) {
  int j = blockIdx.x * 256 + threadIdx.x;          // coalesced across threads
  int p = blockIdx.y;
  float acc = 0.f;
  size_t base = (size_t)p * CR * NN + j;
  for (int r = 0; r < CR; ++r) acc += in[base + (size_t)r * NN];
  part[p * NN + j] = acc;
}

__global__ void k_col_final(const float* __restrict__ part, float* __restrict__ colsum) {
  int j = blockIdx.x * 256 + threadIdx.x;
  float acc = 0.f;
  for (int p = 0; p < CP; ++p) acc += part[p * NN + j];
  colsum[j] = acc;
}

__global__ void k_scalars(const float* __restrict__ rowsum, const float* __restrict__ diag,
                          float* __restrict__ scal) {
  __shared__ float s[512];
  float a = 0.f, b = 0.f;
  for (int k = threadIdx.x; k < NN; k += 256) { a += diag[k]; b += rowsum[k]; }
  s[threadIdx.x] = a; s[256 + threadIdx.x] = b;
  __syncthreads();
  for (int off = 128; off > 0; off >>= 1) {
    if ((int)threadIdx.x < off) {
      s[threadIdx.x] += s[threadIdx.x + off];
      s[256 + threadIdx.x] += s[256 + threadIdx.x + off];
    }
    __syncthreads();
  }
  if (threadIdx.x == 0) { scal[0] = s[0]; scal[1] = s[256]; }
}

// One wave per 16(j) x 16(o) tile at fixed i; D = A(16x8) * B(8x16) via two
// chained V_WMMA_F32_16X16X4_F32, then 8 coalesced NT stores.
__global__ void __launch_bounds__(256)
k_main(const float* __restrict__ in, const float* __restrict__ w,
       const float* __restrict__ rowsum, const float* __restrict__ diag,
       const float* __restrict__ colsum, const float* __restrict__ scal,
       float* __restrict__ out) {
  int lane = threadIdx.x & 31;
  int wid  = threadIdx.x >> 5;
  int tile = blockIdx.x * 8 + wid;
  int i  = tile >> 7;            // 2048 rows
  int j0 = (tile & 127) << 4;    // 128 j-tiles
  int m  = lane & 15;
  bool hi = lane >= 16;
  int j  = j0 + m;

  float tr = scal[0], ts = scal[1];
  float di = diag[i], ri = rowsum[i], ci = colsum[i];

  // Per-lane o-column m of the i-dependent B rows:
  float ceff = di * w[8*NO+m] + ri * w[9*NO+m] + ci * w[10*NO+m]
             + tr * w[13*NO+m] + ts * w[14*NO+m];
  float dg   = di * w[0*NO+m] + ri * w[1*NO+m] + ci * w[2*NO+m]
             + tr * w[3*NO+m] + ts * w[4*NO+m];

  // A: lanes 0-15 = {K0,K1}, lanes 16-31 = {K2,K3}
  v2f a1, a2, b1, b2;
  a1[0] = hi ? diag[j]   : in[(size_t)j * NN + i];   // K0=inputs[j,i]  K2=diag[j]
  a1[1] = hi ? rowsum[j] : in[(size_t)i * NN + j];   // K1=inputs[i,j]  K3=rowsum[j]
  a2[0] = hi ? ((j == i) ? 1.f : 0.f) : colsum[j];   // K0=colsum[j]    K2=(j==i)
  a2[1] = hi ? 0.f : 1.f;                            // K1=1            K3=0

  // B: same half-split, B[k][n] with n = m
  b1[0] = hi ? w[5*NO+m] : w[11*NO+m];               // K0=w11  K2=w5
  b1[1] = hi ? w[6*NO+m] : w[12*NO+m];               // K1=w12  K3=w6
  b2[0] = hi ? dg   : w[7*NO+m];                     // K0=w7   K2=Dg[i]
  b2[1] = hi ? 0.f  : ceff;                          // K1=Ceff K3=0

  v8f c = {};
  c = __builtin_amdgcn_wmma_f32_16x16x4_f32(false, a1, false, b1, (short)0, c, false, false);
  c = __builtin_amdgcn_wmma_f32_16x16x4_f32(false, a2, false, b2, (short)0, c, false, false);

  // D[v][lane]: row j0 + v + (hi?8:0), col m; o is innermost (stride-1).
  size_t ob = ((size_t)i * NN + j0 + (hi ? 8 : 0)) * NO + m;
#pragma unroll
  for (int v = 0; v < 8; ++v)
    __builtin_nontemporal_store(c[v], &out[ob + (size_t)v * NO]);
}

extern "C" void kernel_launch(void* const* d_in, const int* in_sizes, int n_in,
                              void* d_out, int out_size, void* d_ws, size_t ws_size,
                              hipStream_t stream) {
  const float* in = (const float*)d_in[0];
  const float* w  = (const float*)d_in[1];
  float* ws      = (float*)d_ws;
  float* rowsum  = ws;
  float* diag    = ws + 2048;
  float* colsum  = ws + 4096;
  float* scal    = ws + 6144;
  float* part    = ws + 8192;
  float* out     = (float*)d_out;

  k_row_diag<<<NN, 256, 0, stream>>>(in, rowsum, diag);
  k_col_part<<<dim3(NN / 256, CP), 256, 0, stream>>>(in, part);
  k_col_final<<<NN / 256, 256, 0, stream>>>(part, colsum);
  k_scalars<<<1, 256, 0, stream>>>(rowsum, diag, scal);
  k_main<<<(NN * (NN / 16)) / 8, 256, 0, stream>>>(in, w, rowsum, diag, colsum, scal, out);
}